// SparseOutputLoss_15479062134912
// MI455X (gfx1250) — compile-verified
//
#include <hip/hip_runtime.h>
#include <hip/hip_bf16.h>
#include <math.h>

typedef __attribute__((ext_vector_type(2))) float v2f;
typedef __attribute__((ext_vector_type(4))) float f4;
typedef __attribute__((ext_vector_type(8))) float v8f;

// Wave32 full reduction using V_WMMA_F32_16X16X4_F32 as the adder tree.
// Each lane drops its partial into a unique slot of the 16x4 A matrix
// (A-VGPR0 = partial, A-VGPR1 = 0 -> bijection lane->(m,k)).
// B = all-ones => D[m,n] = sum_k A[m,k]; every column of D sums to the
// wave total. Lane L holds 8 M-entries of column N(L); lane L^16 holds the
// other 8 M-entries of the same column -> per-lane sum of the 8 C/D VGPRs
// plus one xor-16 exchange yields the exact f32 wave total in all lanes.
// Requires EXEC == all ones (all callers invoke with full, converged waves).
__device__ __forceinline__ float wave_reduce32(float v) {
#if __has_builtin(__builtin_amdgcn_wmma_f32_16x16x4_f32)
  v2f a; a[0] = v;    a[1] = 0.0f;
  v2f b; b[0] = 1.0f; b[1] = 1.0f;
  v8f c = {};
  c = __builtin_amdgcn_wmma_f32_16x16x4_f32(
      /*neg_a=*/false, a, /*neg_b=*/false, b,
      /*c_mod=*/(short)0, c, /*reuse_a=*/false, /*reuse_b=*/false);
  float t = (c[0] + c[1]) + (c[2] + c[3]) + (c[4] + c[5]) + (c[6] + c[7]);
  t += __shfl_xor(t, 16, 32);
  return t;
#else
  float t = v;
  for (int off = 16; off >= 1; off >>= 1) t += __shfl_xor(t, off, 32);
  return t;
#endif
}

__global__ void sol_init_ws(float* ws) {
  if (threadIdx.x < 9) ws[threadIdx.x] = 0.0f;
}

// Streaming reduction for one pyramid level. Memory-bound: 4 NT b128 loads
// per 4 elements, ~2 FMA + 2 ADD VALU per element.
__global__ void __launch_bounds__(256)
sol_reduce_level(const float* __restrict__ o, const float* __restrict__ om,
                 const float* __restrict__ gt, const float* __restrict__ w,
                 int n4, float* __restrict__ loss_acc,
                 float* __restrict__ w_acc, float* __restrict__ om_acc) {
  const f4* __restrict__ o4  = (const f4*)o;
  const f4* __restrict__ om4 = (const f4*)om;
  const f4* __restrict__ gt4 = (const f4*)gt;
  const f4* __restrict__ w4  = (const f4*)w;

  int tid    = blockIdx.x * blockDim.x + threadIdx.x;
  int stride = gridDim.x * blockDim.x;

  float loss = 0.0f, sw = 0.0f, som = 0.0f;
  for (int i = tid; i < n4; i += stride) {
    f4 ov = __builtin_nontemporal_load(o4  + i);
    f4 mv = __builtin_nontemporal_load(om4 + i);
    f4 gv = __builtin_nontemporal_load(gt4 + i);
    f4 wv = __builtin_nontemporal_load(w4  + i);
#pragma unroll
    for (int k = 0; k < 4; ++k) {
      // om*gt*gt + o*o - 2*om*gt*o == (om*gt - o)^2  because om in {0,1}
      float d = __builtin_fmaf(mv[k], gv[k], -ov[k]);
      loss = __builtin_fmaf(d * d, wv[k], loss);
      sw  += wv[k];
      som += mv[k];
    }
  }

  // Full waves, converged control flow here -> EXEC all ones for WMMA.
  float lw = wave_reduce32(loss);
  float ww = wave_reduce32(sw);
  float mw = wave_reduce32(som);

  __shared__ float s_l[8], s_w[8], s_m[8];
  int lane = threadIdx.x & 31;
  int wid  = threadIdx.x >> 5;
  if (lane == 0) { s_l[wid] = lw; s_w[wid] = ww; s_m[wid] = mw; }
  __syncthreads();
  if (threadIdx.x == 0) {
    int nw = blockDim.x >> 5;
    float L = 0.0f, W = 0.0f, M = 0.0f;
    for (int i = 0; i < nw; ++i) { L += s_l[i]; W += s_w[i]; M += s_m[i]; }
    atomicAdd(loss_acc, L);
    atomicAdd(w_acc, W);
    atomicAdd(om_acc, M);
  }
}

__global__ void sol_finalize(const float* __restrict__ ws, float* __restrict__ out) {
  if (blockIdx.x == 0 && threadIdx.x == 0) {
    float mult = 0.0f;
#pragma unroll
    for (int i = 0; i < 4; ++i)
      mult += ws[1 + i] / fmaxf(ws[5 + i], 1.0f);
    out[0] = sqrtf(ws[0] / mult);
  }
}

extern "C" void kernel_launch(void* const* d_in, const int* in_sizes, int n_in,
                              void* d_out, int out_size, void* d_ws, size_t ws_size,
                              hipStream_t stream) {
  (void)n_in; (void)out_size; (void)ws_size;
  float* ws = (float*)d_ws;

  // ws layout: [0]=total_loss, [1..4]=sum(w) per level, [5..8]=sum(om) per level
  sol_init_ws<<<1, 32, 0, stream>>>(ws);

  for (int lvl = 0; lvl < 4; ++lvl) {
    const float* o  = (const float*)d_in[lvl * 4 + 0];
    const float* om = (const float*)d_in[lvl * 4 + 1];
    const float* gt = (const float*)d_in[lvl * 4 + 2];
    const float* w  = (const float*)d_in[lvl * 4 + 3];
    int n  = in_sizes[lvl * 4 + 0];
    int n4 = n >> 2;  // all level sizes divisible by 4

    int threads = 256;
    int blocks  = (n4 + threads * 8 - 1) / (threads * 8);  // ~8 float4 per thread
    if (blocks < 1) blocks = 1;
    if (blocks > 2048) blocks = 2048;

    sol_reduce_level<<<blocks, threads, 0, stream>>>(
        o, om, gt, w, n4, ws + 0, ws + 1 + lvl, ws + 5 + lvl);
  }

  sol_finalize<<<1, 32, 0, stream>>>(ws, (float*)d_out);
}